// constrained_ouroboros_67405216743850
// MI455X (gfx1250) — compile-verified
//
#include <hip/hip_runtime.h>
#include <math.h>

typedef __attribute__((ext_vector_type(2))) float v2f;
typedef __attribute__((ext_vector_type(8))) float v8f;

#define BB 32
#define LL 8192
#define LZ 8184          // L - 8
#define MMOD 3
#define NLAY 2
#define TCH 128          // timesteps per chunk
#define NCH ((LZ + TCH - 1) / TCH)
#define LOG2E 1.4426950408889634f

__device__ __forceinline__ float siluf(float x) { return x / (1.0f + expf(-x)); }
__device__ __forceinline__ float softplusf(float x) {
    if (x > 20.0f) return x;
    if (x < -20.0f) return expf(x);
    return log1pf(expf(x));
}

// ---------------------------------------------------------------------------
// K1: 9-point derivative stencil -> z[b][t][{x, xdot}]
// ---------------------------------------------------------------------------
__global__ void k_deriv(const float* __restrict__ x, float* __restrict__ z) {
    int idx = blockIdx.x * blockDim.x + threadIdx.x;
    if (idx >= BB * LZ) return;
    int b = idx / LZ, t = idx % LZ;
    const float* xb = x + (size_t)b * LL;
    float xd = (1.0f/280.0f) * xb[t]     + (-4.0f/105.0f) * xb[t+1]
             + (0.2f)        * xb[t+2]   + (-0.8f)        * xb[t+3]
             + (0.8f)        * xb[t+5]   + (-0.2f)        * xb[t+6]
             + (4.0f/105.0f) * xb[t+7]   + (-1.0f/280.0f) * xb[t+8];
    z[(size_t)idx * 2 + 0] = xb[t + 4];
    z[(size_t)idx * 2 + 1] = xd;
}

// ---------------------------------------------------------------------------
// K2: fused 2-layer Mamba per (b, m).  Block = 128 threads (4 waves).
//   Phase A (parallel over t): rmsnorm, WMMA in_proj, conv+silu, WMMA x_proj,
//                              softplus(delta)
//   Phase B (wave 0, serial over t): selective scan, lanes hold (d,n) states
//   Phase C (parallel over t): out_proj + residual
// ---------------------------------------------------------------------------
__global__ __launch_bounds__(128) void k_mamba(
    const float* __restrict__ z,
    const float* __restrict__ in_proj_w,  const float* __restrict__ conv_w,
    const float* __restrict__ conv_b,     const float* __restrict__ x_proj_w,
    const float* __restrict__ dt_proj_w,  const float* __restrict__ dt_proj_b,
    const float* __restrict__ A_log,      const float* __restrict__ D_param,
    const float* __restrict__ norm_w,     const float* __restrict__ out_proj_w,
    float* __restrict__ out_controls)     // (B, LZ, 12), this block fills cols m*4..m*4+3
{
    const int b    = blockIdx.x % BB;
    const int m    = blockIdx.x / BB;
    const int tid  = threadIdx.x;
    const int lane = tid & 31;
    const int wav  = tid >> 5;

    // ---- weights in LDS ----
    __shared__ float s_win[NLAY][32];     // in_proj (8,4)
    __shared__ float s_cw [NLAY][16];     // conv_w (4,4)
    __shared__ float s_cb [NLAY][4];
    __shared__ float s_xp [NLAY][132];    // x_proj (33,4)
    __shared__ float s_dtw[NLAY][4];
    __shared__ float s_dtb[NLAY][4];
    __shared__ float s_Ac [NLAY][64];     // -exp(A_log)*log2e, (4,16)
    __shared__ float s_D  [NLAY][4];
    __shared__ float s_nw [NLAY][4];
    __shared__ float s_wo [NLAY][16];     // out_proj (4,4)
    __shared__ float s_carry[NLAY][12];   // conv history (3 steps x 4 ch)

    // ---- per-chunk activations ----
    __shared__ float s_in [TCH * 4];       // layer input
    __shared__ float s_hn [TCH * 4];       // rmsnorm output
    __shared__ float s_xz [TCH * 16];      // cols 0-3: xc/u, 4-7: silu(res)
    __shared__ float s_pre[(TCH + 3) * 4]; // pre-conv xc with 3-step halo
    __shared__ float s_dbc[TCH * 48];      // 0: dtr, 1-16: B, 17-32: C, 33-36: delta
    __shared__ float s_y  [TCH * 4];       // scan output * silu(res)

    for (int l = 0; l < NLAY; ++l) {
        int base = m * NLAY + l;
        for (int i = tid; i < 32;  i += 128) s_win[l][i] = in_proj_w[base * 32 + i];
        for (int i = tid; i < 16;  i += 128) s_cw [l][i] = conv_w  [base * 16 + i];
        for (int i = tid; i < 132; i += 128) s_xp [l][i] = x_proj_w[base * 132 + i];
        for (int i = tid; i < 64;  i += 128) s_Ac [l][i] = -expf(A_log[base * 64 + i]) * LOG2E;
        for (int i = tid; i < 16;  i += 128) s_wo [l][i] = out_proj_w[base * 16 + i];
        for (int i = tid; i < 4;   i += 128) {
            s_cb [l][i] = conv_b   [base * 4 + i];
            s_dtw[l][i] = dt_proj_w[base * 4 + i];
            s_dtb[l][i] = dt_proj_b[base * 4 + i];
            s_D  [l][i] = D_param  [base * 4 + i];
            s_nw [l][i] = norm_w   [base * 4 + i];
        }
        for (int i = tid; i < 12; i += 128) s_carry[l][i] = 0.0f;
    }
    __syncthreads();

    // scan state (wave 0 lanes): lane -> (d = half, half+2 ; n = lane&15)
    float h_state[NLAY][2] = {{0.f, 0.f}, {0.f, 0.f}};
    float Ac_reg[NLAY][2]  = {{0.f, 0.f}, {0.f, 0.f}};
    if (wav == 0) {
        int half = lane >> 4, n = lane & 15;
#pragma unroll
        for (int l = 0; l < NLAY; ++l) {
            Ac_reg[l][0] = s_Ac[l][half * 16 + n];
            Ac_reg[l][1] = s_Ac[l][(half + 2) * 16 + n];
        }
    }

    const float* zb = z + (size_t)b * LZ * 2;
    const int n16 = lane & 15;
    const int k0  = (lane < 16) ? 0 : 2;
    const int dro = (lane < 16) ? 0 : 8;   // D-matrix row offset for this lane half

    for (int c = 0; c < NCH; ++c) {
        const int t0 = c * TCH;
        const int T  = min(TCH, LZ - t0);

        // x_in[t] = [z[t], z_flipped[t]]
        {
            int i = tid;
            if (i < T) {
                int t = t0 + i, tf = LZ - 1 - t;
                s_in[i*4+0] = zb[(size_t)t  * 2 + 0];
                s_in[i*4+1] = zb[(size_t)t  * 2 + 1];
                s_in[i*4+2] = zb[(size_t)tf * 2 + 0];
                s_in[i*4+3] = zb[(size_t)tf * 2 + 1];
            } else {
                s_in[i*4+0] = s_in[i*4+1] = s_in[i*4+2] = s_in[i*4+3] = 0.f;
            }
        }
        __syncthreads();

#pragma unroll
        for (int l = 0; l < NLAY; ++l) {
            // -------- A1: rmsnorm ; restore conv carry --------
            {
                int i = tid;
                if (i < T) {
                    float a0 = s_in[i*4], a1 = s_in[i*4+1], a2 = s_in[i*4+2], a3 = s_in[i*4+3];
                    float rs = rsqrtf(0.25f * (a0*a0 + a1*a1 + a2*a2 + a3*a3) + 1e-5f);
                    s_hn[i*4+0] = a0 * rs * s_nw[l][0];
                    s_hn[i*4+1] = a1 * rs * s_nw[l][1];
                    s_hn[i*4+2] = a2 * rs * s_nw[l][2];
                    s_hn[i*4+3] = a3 * rs * s_nw[l][3];
                } else {
                    s_hn[i*4+0] = s_hn[i*4+1] = s_hn[i*4+2] = s_hn[i*4+3] = 0.f;
                }
                if (i < 12) s_pre[i] = s_carry[l][i];
            }
            __syncthreads();

            // -------- A2: in_proj via V_WMMA_F32_16X16X4_F32 --------
            // A(16x4): lane half selects K={0,1} or {2,3}; B(4x16): B[k][n]=W[n][k]
            {
                v2f bm;
                bm.x = (n16 < 8) ? s_win[l][n16 * 4 + k0]     : 0.f;
                bm.y = (n16 < 8) ? s_win[l][n16 * 4 + k0 + 1] : 0.f;
#pragma unroll
                for (int tt = 0; tt < 2; ++tt) {
                    int rowbase = (wav * 2 + tt) * 16;
                    int r = rowbase + n16;
                    v2f am; am.x = s_hn[r*4 + k0]; am.y = s_hn[r*4 + k0 + 1];
                    v8f cz = {};
                    v8f dd = __builtin_amdgcn_wmma_f32_16x16x4_f32(
                        false, am, false, bm, (short)0, cz, false, false);
                    int rb = rowbase + dro;
#pragma unroll
                    for (int v = 0; v < 8; ++v) s_xz[(rb + v) * 16 + n16] = dd[v];
                }
            }
            __syncthreads();

            // -------- A3: stage pre-conv xc, silu(res) --------
            {
                int i = tid;
                if (i < T) {
#pragma unroll
                    for (int d = 0; d < 4; ++d) s_pre[(3 + i) * 4 + d] = s_xz[i*16 + d];
#pragma unroll
                    for (int d = 0; d < 4; ++d) s_xz[i*16 + 4 + d] = siluf(s_xz[i*16 + 4 + d]);
                } else {
#pragma unroll
                    for (int d = 0; d < 4; ++d) s_pre[(3 + i) * 4 + d] = 0.f;
                }
            }
            __syncthreads();

            // -------- A4: causal conv (K=4) + silu -> u ; save carry --------
            {
                int i = tid;
                if (i < T) {
#pragma unroll
                    for (int d = 0; d < 4; ++d) {
                        float acc = s_cb[l][d];
#pragma unroll
                        for (int j = 0; j < 4; ++j)
                            acc += s_pre[(i + j) * 4 + d] * s_cw[l][d * 4 + j];
                        s_xz[i*16 + d] = siluf(acc);
                    }
                } else {
#pragma unroll
                    for (int d = 0; d < 4; ++d) s_xz[i*16 + d] = 0.f;
                }
                if (i < 12) s_carry[l][i] = s_pre[(T + i / 4) * 4 + (i % 4)];
            }
            __syncthreads();

            // -------- A5: x_proj (33 cols, 3 N-tiles) via WMMA --------
            {
#pragma unroll
                for (int tt = 0; tt < 2; ++tt) {
                    int rowbase = (wav * 2 + tt) * 16;
                    int r = rowbase + n16;
                    v2f am; am.x = s_xz[r*16 + k0]; am.y = s_xz[r*16 + k0 + 1];
                    int rb = rowbase + dro;
#pragma unroll
                    for (int tn = 0; tn < 3; ++tn) {
                        int ng = tn * 16 + n16;
                        v2f bm;
                        bm.x = (ng < 33) ? s_xp[l][ng * 4 + k0]     : 0.f;
                        bm.y = (ng < 33) ? s_xp[l][ng * 4 + k0 + 1] : 0.f;
                        v8f cz = {};
                        v8f dd = __builtin_amdgcn_wmma_f32_16x16x4_f32(
                            false, am, false, bm, (short)0, cz, false, false);
#pragma unroll
                        for (int v = 0; v < 8; ++v)
                            s_dbc[(rb + v) * 48 + tn * 16 + n16] = dd[v];
                    }
                }
            }
            __syncthreads();

            // -------- A6: delta = softplus(dtr * dt_w + dt_b) --------
            {
                int i = tid;
                if (i < T) {
                    float dtr = s_dbc[i * 48 + 0];
#pragma unroll
                    for (int d = 0; d < 4; ++d)
                        s_dbc[i * 48 + 33 + d] = softplusf(dtr * s_dtw[l][d] + s_dtb[l][d]);
                }
            }
            __syncthreads();

            // -------- B: selective scan (wave 0; 64 states, 2 per lane) --------
            if (wav == 0) {
                const int half = lane >> 4, n = lane & 15;
                float h0 = h_state[l][0], h1 = h_state[l][1];
                const float A0 = Ac_reg[l][0], A1 = Ac_reg[l][1];
                const float D0 = s_D[l][half], D1 = s_D[l][half + 2];
                for (int i = 0; i < T; ++i) {
                    const float* row = &s_dbc[i * 48];
                    float d0 = row[33 + half];
                    float d1 = row[35 + half];
                    float bb = row[1 + n];
                    float cc = row[17 + n];
                    float u0 = s_xz[i*16 + half];
                    float u1 = s_xz[i*16 + half + 2];
                    h0 = exp2f(d0 * A0) * h0 + d0 * bb * u0;
                    h1 = exp2f(d1 * A1) * h1 + d1 * bb * u1;
                    float p = h0 * cc, q = h1 * cc;
#pragma unroll
                    for (int off = 1; off < 16; off <<= 1) {
                        p += __shfl_xor(p, off, 32);
                        q += __shfl_xor(q, off, 32);
                    }
                    if (n == 0) {
                        float sr0 = s_xz[i*16 + 4 + half];
                        float sr1 = s_xz[i*16 + 6 + half];
                        s_y[i*4 + half]     = (p + u0 * D0) * sr0;
                        s_y[i*4 + half + 2] = (q + u1 * D1) * sr1;
                    }
                }
                h_state[l][0] = h0; h_state[l][1] = h1;
            }
            __syncthreads();

            // -------- C: out_proj + residual --------
            {
                int i = tid;
                if (i < T) {
                    float y0 = s_y[i*4], y1 = s_y[i*4+1], y2 = s_y[i*4+2], y3 = s_y[i*4+3];
                    float o[4];
#pragma unroll
                    for (int dp = 0; dp < 4; ++dp)
                        o[dp] = s_in[i*4+dp]
                              + y0 * s_wo[l][dp*4+0] + y1 * s_wo[l][dp*4+1]
                              + y2 * s_wo[l][dp*4+2] + y3 * s_wo[l][dp*4+3];
                    if (l == 0) {
#pragma unroll
                        for (int dp = 0; dp < 4; ++dp) s_in[i*4+dp] = o[dp];
                    } else {
                        float* oc = out_controls + ((size_t)(b * LZ + t0 + i)) * 12 + m * 4;
#pragma unroll
                        for (int dp = 0; dp < 4; ++dp) oc[dp] = o[dp];
                    }
                }
            }
            __syncthreads();
        } // layer
    } // chunk
}

// ---------------------------------------------------------------------------
// K3: heads -> raw |omega|, gamma/TAU, relu(d)
// ---------------------------------------------------------------------------
__global__ void k_head(const float* __restrict__ ctrl,
                       const float* __restrict__ omega_w, const float* __restrict__ omega_b,
                       const float* __restrict__ gamma_w, const float* __restrict__ gamma_b,
                       const float* __restrict__ d_w,     const float* __restrict__ d_b,
                       float* __restrict__ w_om, float* __restrict__ w_ga,
                       float* __restrict__ w_dd) {
    int idx = blockIdx.x * blockDim.x + threadIdx.x;
    if (idx >= BB * LZ) return;
    const float* cr = ctrl + (size_t)idx * 12;
    float om = omega_b[0], ga = gamma_b[0], dd = d_b[0];
#pragma unroll
    for (int k = 0; k < 4; ++k) {
        om += cr[k]     * omega_w[k];
        ga += cr[4 + k] * gamma_w[k];
        dd += cr[8 + k] * d_w[k];
    }
    w_om[idx] = fabsf(om);
    w_ga[idx] = ga * 0.001f;
    w_dd[idx] = fmaxf(dd, 0.0f);
}

// ---------------------------------------------------------------------------
// K4: width-10 box smooth (window [t-4, t+5], zero-padded) + yhat
// ---------------------------------------------------------------------------
__global__ void k_smooth_yhat(const float* __restrict__ w_om, const float* __restrict__ w_ga,
                              const float* __restrict__ w_dd, const float* __restrict__ z,
                              const float* __restrict__ b_param,
                              float* __restrict__ yhat, float* __restrict__ s_om,
                              float* __restrict__ s_ga, float* __restrict__ s_dd) {
    int idx = blockIdx.x * blockDim.x + threadIdx.x;
    if (idx >= BB * LZ) return;
    int b = idx / LZ, t = idx % LZ;
    const float* po = w_om + (size_t)b * LZ;
    const float* pg = w_ga + (size_t)b * LZ;
    const float* pd = w_dd + (size_t)b * LZ;
    float so = 0.f, sg = 0.f, sd = 0.f;
    int lo = t - 4; if (lo < 0) lo = 0;
    int hi = t + 5; if (hi > LZ - 1) hi = LZ - 1;
    for (int tt = lo; tt <= hi; ++tt) { so += po[tt]; sg += pg[tt]; sd += pd[tt]; }
    so *= 0.1f; sg *= 0.1f; sd *= 0.1f;
    s_om[idx] = so; s_ga[idx] = sg; s_dd[idx] = sd;

    float z1 = z[(size_t)idx * 2 + 0];
    float z2 = z[(size_t)idx * 2 + 1] * 10000.0f;      // / DT
    float bc = fmaxf(b_param[0], 0.0f) * 0.001f;       // relu(b)/TAU
    yhat[idx] = -so * so * z1 + sg * z2 - bc * z1 * z1 * z2 - sd;
}

// ---------------------------------------------------------------------------
// K5: per-(signal,b) variance of diff(smoothed)[1:], ddof=1 (deterministic tree)
// ---------------------------------------------------------------------------
__global__ void k_pen_partial(const float* __restrict__ s_om, const float* __restrict__ s_ga,
                              const float* __restrict__ s_dd, float* __restrict__ vvar) {
    int s = blockIdx.x / BB;
    int b = blockIdx.x % BB;
    const float* p = (s == 0 ? s_om : (s == 1 ? s_ga : s_dd)) + (size_t)b * LZ;
    __shared__ double r1[256], r2[256];
    double a1 = 0.0, a2 = 0.0;
    for (int i = 1 + (int)threadIdx.x; i <= LZ - 2; i += 256) {
        float g = p[i + 1] - p[i];
        a1 += (double)g;
        a2 += (double)g * (double)g;
    }
    r1[threadIdx.x] = a1; r2[threadIdx.x] = a2;
    __syncthreads();
    for (int ofs = 128; ofs > 0; ofs >>= 1) {
        if ((int)threadIdx.x < ofs) {
            r1[threadIdx.x] += r1[threadIdx.x + ofs];
            r2[threadIdx.x] += r2[threadIdx.x + ofs];
        }
        __syncthreads();
    }
    if (threadIdx.x == 0) {
        double n = (double)(LZ - 2);
        vvar[blockIdx.x] = (float)((r2[0] - r1[0] * r1[0] / n) / (n - 1.0));
    }
}

__global__ void k_pen_final(const float* __restrict__ vvar, float* __restrict__ pen) {
    if (blockIdx.x == 0 && threadIdx.x == 0) {
        float acc = 0.f;
        for (int s = 0; s < 3; ++s) {
            float mm = 0.f;
            for (int b = 0; b < BB; ++b) mm += vvar[s * BB + b];
            acc += mm / (float)BB;
        }
        pen[0] = acc / 3.0f;
    }
}

// ---------------------------------------------------------------------------
extern "C" void kernel_launch(void* const* d_in, const int* in_sizes, int n_in,
                              void* d_out, int out_size, void* d_ws, size_t ws_size,
                              hipStream_t stream) {
    (void)in_sizes; (void)n_in; (void)out_size; (void)ws_size;
    const float* x         = (const float*)d_in[0];
    const float* in_proj_w = (const float*)d_in[1];
    const float* conv_w    = (const float*)d_in[2];
    const float* conv_b    = (const float*)d_in[3];
    const float* x_proj_w  = (const float*)d_in[4];
    const float* dt_proj_w = (const float*)d_in[5];
    const float* dt_proj_b = (const float*)d_in[6];
    const float* A_log     = (const float*)d_in[7];
    const float* D_param   = (const float*)d_in[8];
    const float* norm_w    = (const float*)d_in[9];
    const float* out_proj_w= (const float*)d_in[10];
    const float* omega_w   = (const float*)d_in[11];
    const float* omega_b   = (const float*)d_in[12];
    const float* gamma_w   = (const float*)d_in[13];
    const float* gamma_b   = (const float*)d_in[14];
    const float* d_w       = (const float*)d_in[15];
    const float* d_b       = (const float*)d_in[16];
    const float* b_param   = (const float*)d_in[17];

    float* out = (float*)d_out;
    float* ws  = (float*)d_ws;
    const size_t NE = (size_t)BB * LZ;

    float* z    = ws;               // NE*2
    float* w_om = ws + NE * 2;      // NE
    float* w_ga = w_om + NE;
    float* w_dd = w_ga + NE;
    float* sm_o = w_dd + NE;
    float* sm_g = sm_o + NE;
    float* sm_d = sm_g + NE;
    float* vvar = sm_d + NE;        // 96

    float* yhat = out;              // (B, LZ, 1)
    float* ctrl = out + NE;         // (B, LZ, 12)
    float* pen  = out + NE * 13;    // scalar

    const int thr = 256;
    const int blocks = (int)((NE + thr - 1) / thr);

    k_deriv<<<blocks, thr, 0, stream>>>(x, z);
    k_mamba<<<BB * MMOD, 128, 0, stream>>>(z, in_proj_w, conv_w, conv_b, x_proj_w,
                                           dt_proj_w, dt_proj_b, A_log, D_param,
                                           norm_w, out_proj_w, ctrl);
    k_head<<<blocks, thr, 0, stream>>>(ctrl, omega_w, omega_b, gamma_w, gamma_b,
                                       d_w, d_b, w_om, w_ga, w_dd);
    k_smooth_yhat<<<blocks, thr, 0, stream>>>(w_om, w_ga, w_dd, z, b_param,
                                              yhat, sm_o, sm_g, sm_d);
    k_pen_partial<<<MMOD * BB, 256, 0, stream>>>(sm_o, sm_g, sm_d, vvar);
    k_pen_final<<<1, 32, 0, stream>>>(vvar, pen);
}